// GlobalModel_45990509805614
// MI455X (gfx1250) — compile-verified
//
#include <hip/hip_runtime.h>
#include <hip/hip_bf16.h>

// ---------------------------------------------------------------------------
// GlobalModel: segment-mean (sorted batch) + concat + 2-layer MLP (ReLU)
//   Phase 1: HBM-bound streaming reduction (~2.05 GB ~= 88us @ 23.3 TB/s),
//            float4-vectorized, fully coalesced.
//   Phase 2: L2-resident GEMMs on V_WMMA_F32_16X16X4_F32 (exact f32).
//            16(M)x64(N) tile per wave; K processed in super-steps of 16:
//            all 20 fragment loads issued before the 16 WMMAs so the
//            scheduler can clause loads and stagger s_wait_loadcnt.
// ---------------------------------------------------------------------------

typedef __attribute__((ext_vector_type(2))) float v2f;
typedef __attribute__((ext_vector_type(8))) float v8f;

// ---- Kernel 1: segment boundaries via binary search on sorted batch --------
__global__ void seg_bounds_kernel(const long long* __restrict__ batch,
                                  int* __restrict__ seg, int n, int b) {
    int g = blockIdx.x * blockDim.x + threadIdx.x;
    if (g > b) return;
    long long key = (long long)g;
    int lo = 0, hi = n;
    while (lo < hi) {
        int mid = (lo + hi) >> 1;
        if (batch[mid] < key) lo = mid + 1; else hi = mid;
    }
    seg[g] = lo;  // first row with batch >= g
}

// ---- Kernel 2: per-graph mean of node/edge features + concat with u --------
// grid.x = B, block = 128. Lanes 0..63 own node cols [4t,4t+4), lanes 64..127
// own edge cols. float4 loads => global_load_b128, fully coalesced per row.
__global__ void __launch_bounds__(128)
seg_mean_concat_kernel(const float* __restrict__ node,
                       const float* __restrict__ edge,
                       const float* __restrict__ u,
                       const int* __restrict__ seg,
                       float* __restrict__ cat) {
    const int g = blockIdx.x;
    const int t = threadIdx.x;              // 0..127
    const int s = seg[g];
    const int e = seg[g + 1];

    const int  c4     = (t & 63) * 4;
    const bool isNode = (t < 64);
    const float* src = (isNode ? node : edge) + (size_t)s * 256 + c4;

    float ax = 0.f, ay = 0.f, az = 0.f, aw = 0.f;
    for (int r = s; r < e; ++r) {
        float4 v = *(const float4*)src;
        ax += v.x; ay += v.y; az += v.z; aw += v.w;
        src += 256;
    }
    const int cnt = e - s;
    const float inv = 1.0f / (float)(cnt > 0 ? cnt : 1);

    float* row = cat + (size_t)g * 640 + (isNode ? 0 : 256) + c4;
    float4 res; res.x = ax * inv; res.y = ay * inv; res.z = az * inv; res.w = aw * inv;
    *(float4*)row = res;

    if (t < 32) {   // copy u[g, 0:128] into concat cols 512..639
        const float4* up = (const float4*)(u + (size_t)g * 128);
        ((float4*)(cat + (size_t)g * 640 + 512))[t] = up[t];
    }
}

// ---- Kernel 3: C[M,N] = relu(A[M,K] @ W[N,K]^T + bias[N]) via f32 WMMA -----
// f32 16x4 A-frag layout: lane L holds contiguous {K=k0,K=k0+1} of row
// (L&15), k0 = 2*(L>>4); B-frag mirrors with N=(L&15). Each fragment is one
// contiguous float2 (global_load_b64). One wave owns a 16x64 tile: per K=16
// super-step it loads 4 A frags + 4x4 W frags first, then runs 16 WMMAs.
__global__ void __launch_bounds__(256)
gemm_bias_relu_f32wmma(const float* __restrict__ A,
                       const float* __restrict__ W,
                       const float* __restrict__ bias,
                       float* __restrict__ C,
                       int M, int N, int K) {
    const int wave = blockIdx.x * (blockDim.x >> 5) + (threadIdx.x >> 5);
    const int lane = threadIdx.x & 31;

    const int tilesN = N >> 6;                  // 64-wide N tiles
    const int tM = wave / tilesN;
    const int tN = (wave - tM * tilesN) * 4;    // first 16-wide N subtile
    if (tM * 16 >= M) return;                   // wave-uniform: EXEC stays all-1s

    const int r  = lane & 15;
    const int hk = (lane >> 4) * 2;

    const float* ap = A + (size_t)(tM * 16 + r) * K + hk;
    const float* wp0 = W + (size_t)((tN + 0) * 16 + r) * K + hk;
    const float* wp1 = W + (size_t)((tN + 1) * 16 + r) * K + hk;
    const float* wp2 = W + (size_t)((tN + 2) * 16 + r) * K + hk;
    const float* wp3 = W + (size_t)((tN + 3) * 16 + r) * K + hk;

    v8f acc0 = (v8f){0.f,0.f,0.f,0.f,0.f,0.f,0.f,0.f};
    v8f acc1 = acc0, acc2 = acc0, acc3 = acc0;

    // K super-steps of 16 (K is a multiple of 16 for both layers)
    for (int k = 0; k < K; k += 16) {
        // ---- issue all 20 independent loads first ----
        v2f av[4], w0v[4], w1v[4], w2v[4], w3v[4];
        #pragma unroll
        for (int kk = 0; kk < 4; ++kk) av[kk]  = *(const v2f*)(ap  + 4 * kk);
        #pragma unroll
        for (int kk = 0; kk < 4; ++kk) w0v[kk] = *(const v2f*)(wp0 + 4 * kk);
        #pragma unroll
        for (int kk = 0; kk < 4; ++kk) w1v[kk] = *(const v2f*)(wp1 + 4 * kk);
        #pragma unroll
        for (int kk = 0; kk < 4; ++kk) w2v[kk] = *(const v2f*)(wp2 + 4 * kk);
        #pragma unroll
        for (int kk = 0; kk < 4; ++kk) w3v[kk] = *(const v2f*)(wp3 + 4 * kk);

        // ---- then 16 WMMAs (A fragment reused 4x per kk) ----
        #pragma unroll
        for (int kk = 0; kk < 4; ++kk) {
            acc0 = __builtin_amdgcn_wmma_f32_16x16x4_f32(false, av[kk], false, w0v[kk], (short)0, acc0, false, false);
            acc1 = __builtin_amdgcn_wmma_f32_16x16x4_f32(false, av[kk], false, w1v[kk], (short)0, acc1, false, false);
            acc2 = __builtin_amdgcn_wmma_f32_16x16x4_f32(false, av[kk], false, w2v[kk], (short)0, acc2, false, false);
            acc3 = __builtin_amdgcn_wmma_f32_16x16x4_f32(false, av[kk], false, w3v[kk], (short)0, acc3, false, false);
        }

        ap += 16; wp0 += 16; wp1 += 16; wp2 += 16; wp3 += 16;
    }

    // C/D layout: VGPR i -> M = i + 8*(lane>=16), N = lane&15
    const int m0 = tM * 16 + (lane >> 4) * 8;
    v8f accs[4] = {acc0, acc1, acc2, acc3};
    #pragma unroll
    for (int j = 0; j < 4; ++j) {
        const int n = (tN + j) * 16 + r;
        const float bb = bias[n];
        #pragma unroll
        for (int i = 0; i < 8; ++i) {
            float v = accs[j][i] + bb;
            C[(size_t)(m0 + i) * N + n] = v > 0.0f ? v : 0.0f;
        }
    }
}

// ---------------------------------------------------------------------------
extern "C" void kernel_launch(void* const* d_in, const int* in_sizes, int n_in,
                              void* d_out, int out_size, void* d_ws, size_t ws_size,
                              hipStream_t stream) {
    const float*     node  = (const float*)d_in[0];     // [N,256]
    const float*     edge  = (const float*)d_in[1];     // [N,256]
    const float*     u     = (const float*)d_in[2];     // [B,128]
    const long long* batch = (const long long*)d_in[3]; // [N] int64, sorted
    const float*     W1    = (const float*)d_in[4];     // [512,640]
    const float*     b1    = (const float*)d_in[5];     // [512]
    const float*     W2    = (const float*)d_in[6];     // [128,512]
    const float*     b2    = (const float*)d_in[7];     // [128]
    float*           out   = (float*)d_out;             // [B,128]

    const int N = in_sizes[0] / 256;                    // 1,000,000
    const int B = in_sizes[2] / 128;                    // 8192
    const int HID = 512, DCAT = 640, DOUT = 128;

    // workspace layout
    char* ws = (char*)d_ws;
    int*   seg = (int*)ws;                                         // [B+1]
    size_t off = ((size_t)(B + 1) * sizeof(int) + 255) & ~(size_t)255;
    float* cat = (float*)(ws + off);                               // [B,640]
    off += (size_t)B * DCAT * sizeof(float);
    float* h   = (float*)(ws + off);                               // [B,512]

    // 1) segment boundaries
    seg_bounds_kernel<<<(B + 1 + 255) / 256, 256, 0, stream>>>(batch, seg, N, B);

    // 2) segment means + concat (HBM-bound phase)
    seg_mean_concat_kernel<<<B, 128, 0, stream>>>(node, edge, u, seg, cat);

    // 3) h = relu(cat @ W1^T + b1)   [B,640]x[512,640]^T -> [B,512]
    {
        int waves = (B / 16) * (HID / 64);              // 512*8 = 4096
        gemm_bias_relu_f32wmma<<<waves / 8, 256, 0, stream>>>(cat, W1, b1, h,
                                                              B, HID, DCAT);
    }
    // 4) out = relu(h @ W2^T + b2)   [B,512]x[128,512]^T -> [B,128]
    {
        int waves = (B / 16) * (DOUT / 64);             // 512*2 = 1024
        gemm_bias_relu_f32wmma<<<waves / 8, 256, 0, stream>>>(h, W2, b2, out,
                                                              B, DOUT, HID);
    }
}